// OrientedRepPointsLoss_86174223827190
// MI455X (gfx1250) — compile-verified
//
#include <hip/hip_runtime.h>
#include <stdint.h>

// CDNA5 / gfx1250, wave32.
// Strategy: express the (point - gt_center)/gt_wh squared distance as a K=4
// quadratic form and evaluate 16gt x 16pt tiles with V_WMMA_F32_16X16X4_F32.
//   d2 = iw2*px^2 + ih2*py^2 + (-2*iw2*gx)*px + (-2*ih2*gy)*py + (iw2*gx^2+ih2*gy^2)
// A row (per GT, K=0..3)  : [iw2, ih2, -2*iw2*gx, -2*ih2*gy]
// B col (per point, K=0..3): [px^2, py^2, px, py]
// C (accumulator init)     : iw2*gx^2 + ih2*gy^2  (broadcast across columns)
// Level mask applied post-WMMA; per-row running min kept as packed u64
// (d2_bits<<32 | point_idx) which is order-correct for d2 >= 0 and breaks
// ties toward the smaller index (== jnp.argmin first-hit semantics).

typedef __attribute__((ext_vector_type(2))) float v2f;
typedef __attribute__((ext_vector_type(8))) float v8f;

#define INF_BITS 0x7F800000u

// ---------------- Kernel A: per-GT parameters ----------------
__global__ void gt_params_kernel(const float* __restrict__ gt, int K, int Kp,
                                 float* __restrict__ gx, float* __restrict__ gy,
                                 float* __restrict__ iw2, float* __restrict__ ih2,
                                 int* __restrict__ glvl,
                                 unsigned long long* __restrict__ cand,
                                 int lvl_min, int lvl_max) {
  int k = blockIdx.x * blockDim.x + threadIdx.x;
  if (k >= Kp) return;
  cand[k] = ~0ull;
  if (k >= K) { gx[k] = 0.f; gy[k] = 0.f; iw2[k] = 0.f; ih2[k] = 0.f; glvl[k] = -1; return; }
  const float* g = gt + (size_t)k * 8;
  float xmn = g[0], xmx = g[0], ymn = g[1], ymx = g[1];
#pragma unroll
  for (int i = 1; i < 4; ++i) {
    float x = g[2 * i], y = g[2 * i + 1];
    xmn = fminf(xmn, x); xmx = fmaxf(xmx, x);
    ymn = fminf(ymn, y); ymx = fmaxf(ymx, y);
  }
  float cx = (xmn + xmx) * 0.5f, cy = (ymn + ymx) * 0.5f;
  float w = fmaxf(xmx - xmn, 1e-6f), h = fmaxf(ymx - ymn, 1e-6f);
  // SCALE = 4 -> w/SCALE == w*0.25 ; int cast truncates like .astype(int32)
  int lvl = (int)((log2f(w * 0.25f) + log2f(h * 0.25f)) * 0.5f);
  lvl = min(max(lvl, lvl_min), lvl_max);
  float iw = 1.0f / w, ih = 1.0f / h;
  gx[k] = cx; gy[k] = cy; iw2[k] = iw * iw; ih2[k] = ih * ih; glvl[k] = lvl;
}

// ---------------- Kernel B: init per-point state ----------------
__global__ void init_kernel(const float* __restrict__ pts, int N,
                            unsigned* __restrict__ best_bits, int* __restrict__ a,
                            int* __restrict__ plvl, int BIG) {
  int n = blockIdx.x * blockDim.x + threadIdx.x;
  if (n >= N) return;
  best_bits[n] = INF_BITS;
  a[n] = BIG;
  // strides are exact powers of two -> exponent extraction == log2
  plvl[n] = (int)(__float_as_uint(pts[(size_t)n * 3 + 2]) >> 23) - 127;
}

// ---------------- Kernel C: WMMA distance tiles + per-GT min ----------------
// TAIL=false is the fast path (N % 16 == 0): no clamps, no exec-mask churn in
// the WMMA loop; TAIL=true is the fully general fallback.
template <bool TAIL>
__global__ void __launch_bounds__(256) cand_kernel(
    const float* __restrict__ pts, int N, int numTiles,
    const float* __restrict__ gx, const float* __restrict__ gy,
    const float* __restrict__ iw2a, const float* __restrict__ ih2a,
    const int* __restrict__ glvl, const int* __restrict__ plvl,
    unsigned long long* __restrict__ cand) {
  const int lane = threadIdx.x & 31;
  const int wave = threadIdx.x >> 5;
  const int k0 = blockIdx.y * 16;
  const int col = lane & 15;
  const bool hiHalf = lane >= 16;

  // A operand: lanes 0-15 hold (K=0,K=1) for M=lane; lanes 16-31 hold (K=2,K=3) for M=lane-16
  const int ka = k0 + col;
  const float iw2v = iw2a[ka], ih2v = ih2a[ka];
  v2f A;
  if (!hiHalf) { A.x = iw2v;                  A.y = ih2v; }
  else         { A.x = -2.0f * iw2v * gx[ka]; A.y = -2.0f * ih2v * gy[ka]; }

  // C operand + row levels: VGPR v holds row (k0+v) for lanes 0-15, row (k0+v+8) for lanes 16-31
  const int rbase = k0 + (hiHalf ? 8 : 0);
  v8f C;
  int rowlvl[8];
#pragma unroll
  for (int v = 0; v < 8; ++v) {
    int kr = rbase + v;
    float gxx = gx[kr], gyy = gy[kr];
    C[v] = iw2a[kr] * gxx * gxx + ih2a[kr] * gyy * gyy;
    rowlvl[v] = glvl[kr];
  }

  unsigned long long mp[8];
#pragma unroll
  for (int v = 0; v < 8; ++v) mp[v] = ~0ull;

  const int totalWaves = gridDim.x * (blockDim.x >> 5);
#pragma unroll 2
  for (int t = blockIdx.x * (blockDim.x >> 5) + wave; t < numTiles; t += totalWaves) {
    const int n = t * 16 + col;
    int nc = n, pl;
    if (TAIL) {
      nc = min(n, N - 1);
      pl = (n < N) ? plvl[nc] : -2;
    } else {
      pl = plvl[n];
    }
    const float px = pts[(size_t)nc * 3 + 0];
    const float py = pts[(size_t)nc * 3 + 1];

    // B operand: lanes 0-15 hold (K=0,K=1) for N=col; lanes 16-31 hold (K=2,K=3)
    v2f B;
    if (!hiHalf) { B.x = px * px; B.y = py * py; }
    else         { B.x = px;      B.y = py; }

    v8f D = __builtin_amdgcn_wmma_f32_16x16x4_f32(
        /*neg_a=*/false, A, /*neg_b=*/false, B,
        /*c_mod=*/(short)0, C, /*reuse_a=*/false, /*reuse_b=*/false);

#pragma unroll
    for (int v = 0; v < 8; ++v) {
      float d2 = (pl == rowlvl[v]) ? fmaxf(D[v], 0.0f) : __int_as_float(0x7F800000);
      unsigned long long p =
          ((unsigned long long)__float_as_uint(d2) << 32) | (unsigned)n;
      mp[v] = (p < mp[v]) ? p : mp[v];
    }
  }

  // reduce across the 16 columns (xor masks 1,2,4,8 stay within each 16-lane half)
#pragma unroll
  for (int v = 0; v < 8; ++v) {
    unsigned long long m = mp[v];
#pragma unroll
    for (int off = 1; off < 16; off <<= 1) {
      unsigned long long o = __shfl_xor(m, off, 32);
      m = (o < m) ? o : m;
    }
    if (col == 0) atomicMin(&cand[rbase + v], m);
  }
}

// ---------------- Kernel D: best[n] = scatter-min of cand distances ----------------
__global__ void scatter_best_kernel(const unsigned long long* __restrict__ cand, int K,
                                    unsigned* __restrict__ best_bits) {
  int k = blockIdx.x * blockDim.x + threadIdx.x;
  if (k >= K) return;
  unsigned long long c = cand[k];
  unsigned hb = (unsigned)(c >> 32);
  if (hb < INF_BITS) atomicMin(&best_bits[(unsigned)c], hb);
}

// ---------------- Kernel E: a[n] = scatter-min of (gt_id+1) among exact winners ----------------
__global__ void scatter_key_kernel(const unsigned long long* __restrict__ cand, int K,
                                   const unsigned* __restrict__ best_bits,
                                   int* __restrict__ a) {
  int k = blockIdx.x * blockDim.x + threadIdx.x;
  if (k >= K) return;
  unsigned long long c = cand[k];
  unsigned hb = (unsigned)(c >> 32);
  unsigned idx = (unsigned)c;
  if (hb < INF_BITS && best_bits[idx] == hb) atomicMin(&a[idx], k + 1);
}

// ---------------- Kernel F: finalize outputs ----------------
__global__ void finalize_kernel(int N, int BIG, const int* __restrict__ gt_labels,
                                float* __restrict__ out) {
  int n = blockIdx.x * blockDim.x + threadIdx.x;
  if (n >= N) return;
  int* a = (int*)out;                               // aliases out[0..N)
  unsigned* bb = (unsigned*)(out + 2 * (size_t)N);  // aliases out[2N..3N)
  int av = a[n];
  int inds = (av >= BIG) ? 0 : av;
  int lab = (inds > 0) ? gt_labels[inds - 1] : 0;
  float dist = sqrtf(__uint_as_float(bb[n]));       // inf stays inf
  out[n] = (float)inds;
  out[(size_t)N + n] = (float)lab;
  out[2 * (size_t)N + n] = dist;
}

extern "C" void kernel_launch(void* const* d_in, const int* in_sizes, int n_in,
                              void* d_out, int out_size, void* d_ws, size_t ws_size,
                              hipStream_t stream) {
  const float* points = (const float*)d_in[0];
  const float* gt = (const float*)d_in[1];
  const int* gt_labels = (const int*)d_in[2];

  const int N = in_sizes[0] / 3;
  const int K = in_sizes[1] / 8;
  const int Kp = ((K + 15) / 16) * 16;
  const int BIG = K + 1;
  float* out = (float*)d_out;

  // tiny workspace: per-GT params + packed candidates (~6 KB)
  float* gx = (float*)d_ws;
  float* gy = gx + Kp;
  float* iw2 = gy + Kp;
  float* ih2 = iw2 + Kp;
  int* glvl = (int*)(ih2 + Kp);
  unsigned long long* cand = (unsigned long long*)(glvl + Kp);  // Kp even -> 8B aligned

  // N-sized intermediates staged inside d_out (rewritten in-place by finalize)
  int* a = (int*)out;
  int* plvl = (int*)(out + (size_t)N);
  unsigned* best_bits = (unsigned*)(out + 2 * (size_t)N);

  // lvl_min/lvl_max follow from STRIDES = [8..128] -> log2 in [3,7]
  gt_params_kernel<<<(Kp + 255) / 256, 256, 0, stream>>>(
      gt, K, Kp, gx, gy, iw2, ih2, glvl, cand, 3, 7);

  init_kernel<<<(N + 255) / 256, 256, 0, stream>>>(points, N, best_bits, a, plvl, BIG);

  const int numTiles = (N + 15) / 16;
  dim3 gridC(64, Kp / 16);
  if ((N & 15) == 0) {
    cand_kernel<false><<<gridC, 256, 0, stream>>>(points, N, numTiles, gx, gy,
                                                  iw2, ih2, glvl, plvl, cand);
  } else {
    cand_kernel<true><<<gridC, 256, 0, stream>>>(points, N, numTiles, gx, gy,
                                                 iw2, ih2, glvl, plvl, cand);
  }

  scatter_best_kernel<<<(K + 255) / 256, 256, 0, stream>>>(cand, K, best_bits);
  scatter_key_kernel<<<(K + 255) / 256, 256, 0, stream>>>(cand, K, best_bits, a);
  finalize_kernel<<<(N + 255) / 256, 256, 0, stream>>>(N, BIG, gt_labels, out);
}